// SpectralConv2d_14602888806545
// MI455X (gfx1250) — compile-verified
//
#include <hip/hip_runtime.h>
#include <math.h>

typedef __attribute__((ext_vector_type(2))) float v2f;
typedef __attribute__((ext_vector_type(8))) float v8f;

#define TWO_PI_OVER_256 0.0245436926061702596f

__device__ __forceinline__ v8f wmma_f32(v2f a, v2f b, v8f c) {
  // D = A(16x4) * B(4x16) + C, fp32, wave32
  return __builtin_amdgcn_wmma_f32_16x16x4_f32(false, a, false, b, (short)0, c,
                                               false, false);
}

__device__ __forceinline__ v8f v8f_zero() {
  v8f z = {0.f, 0.f, 0.f, 0.f, 0.f, 0.f, 0.f, 0.f};
  return z;
}

// 32-bit LDS byte address of a __shared__ object (addrspace(3) ptrs are 32-bit)
__device__ __forceinline__ unsigned lds_off(void* p) {
  return (unsigned)(unsigned long long)(__attribute__((address_space(3))) char*)p;
}

// CDNA5 async memory->LDS copies (ASYNCcnt); per-lane LDS dest + global src.
__device__ __forceinline__ void async_ld_b128(unsigned lds_byte, const void* g) {
  asm volatile("global_load_async_to_lds_b128 %0, %1, off"
               :
               : "v"(lds_byte), "v"((unsigned long long)g)
               : "memory");
}
__device__ __forceinline__ void async_ld_b32(unsigned lds_byte, const void* g) {
  asm volatile("global_load_async_to_lds_b32 %0, %1, off"
               :
               : "v"(lds_byte), "v"((unsigned long long)g)
               : "memory");
}
__device__ __forceinline__ void wait_async0() {
  asm volatile("s_wait_asynccnt 0" ::: "memory");
}

// ---------------------------------------------------------------------------
// Kernel 1: partial row DFT (real -> 32 complex modes), as GEMM.
// A: x rows (M = 262144, K = 256), B: [cos | -sin] twiddles (256 x 64).
// Xrow[row][0..31] = Re, Xrow[row][32..63] = Im. Unscaled (1/256 in kernel 2).
// ---------------------------------------------------------------------------
__global__ __launch_bounds__(256) void k_row_dft(const float* __restrict__ x,
                                                 float* __restrict__ Xrow) {
  __shared__ __align__(16) float As[32 * 260];  // 32 rows, padded stride 260
  const int tid = threadIdx.x;
  const int wave = tid >> 5, lane = tid & 31;
  const int half = lane >> 4, l = lane & 15;
  const long rowBase = (long)blockIdx.x * 32;

  const float* src = x + rowBase * 256;
  for (int idx = tid * 4; idx < 32 * 256; idx += 1024) {
    const int r = idx >> 8, c = idx & 255;
    async_ld_b128(lds_off(&As[r * 260 + c]), src + idx);
  }
  wait_async0();
  __syncthreads();

  const int mtile = wave >> 2;       // 0..1 (16 rows each)
  const int ntile = wave & 3;        // 0..3
  const int n = ntile * 16 + l;      // output column 0..63
  const int kx = n & 31;
  const bool isCos = (n < 32);

  // z = exp(-i*2pi*kx*v/256); v starts at 2*half (+1 for .y), step 4
  const float phi = -TWO_PI_OVER_256 * (float)kx;
  float c0, s0, c1, s1, cs, ss;
  __sincosf(phi * (float)(2 * half), &s0, &c0);
  __sincosf(phi * (float)(2 * half + 1), &s1, &c1);
  __sincosf(phi * 4.0f, &ss, &cs);

  v8f acc = v8f_zero();
  const float* Arow = &As[(mtile * 16 + l) * 260 + 2 * half];
  for (int k0 = 0; k0 < 256; k0 += 4) {
    v2f a; a.x = Arow[k0]; a.y = Arow[k0 + 1];
    v2f b; b.x = isCos ? c0 : s0; b.y = isCos ? c1 : s1;
    acc = wmma_f32(a, b, acc);
    float t;
    t = c0 * cs - s0 * ss; s0 = c0 * ss + s0 * cs; c0 = t;
    t = c1 * cs - s1 * ss; s1 = c1 * ss + s1 * cs; c1 = t;
  }

  float* out = Xrow + (rowBase + mtile * 16 + 8 * half) * 64 + n;
#pragma unroll
  for (int jj = 0; jj < 8; ++jj) out[(long)jj * 64] = acc[jj];
}

// ---------------------------------------------------------------------------
// Kernel 2: column DFT 256 -> 63 shifted modes (k = j-31), complex.
// Per (b,i): P = Tr*[Xr|Xi], Q = Ti*[Xr|Xi]; outR = TrXr - TiXi, outI = TrXi + TiXr.
// Twiddles on A side via recurrence; forward ortho scale 1/256 folded in.
// ---------------------------------------------------------------------------
__global__ __launch_bounds__(256) void k_col_dft(const float* __restrict__ Xrow,
                                                 float* __restrict__ xftR,
                                                 float* __restrict__ xftI) {
  __shared__ __align__(16) float Xs[256 * 64];  // 64 KB slice for this (b,i)
  const int tid = threadIdx.x;
  const int wave = tid >> 5, lane = tid & 31;
  const int half = lane >> 4, l = lane & 15;
  const int bi = blockIdx.x;  // b*64 + i

  const float* src = Xrow + (long)bi * 256 * 64;
  for (int idx = tid * 4; idx < 256 * 64; idx += 1024)
    async_ld_b128(lds_off(&Xs[idx]), src + idx);
  wait_async0();
  __syncthreads();

  // 16 output tiles: mtile(4) x part(R/I) x nhalf(2); 8 waves -> 2 each
  for (int t = wave; t < 16; t += 8) {
    const int mtile = t >> 2;
    const int part = (t >> 1) & 1;  // 0 = real out, 1 = imag out
    const int nh = t & 1;
    const int j = mtile * 16 + l;   // mode row (j=63 discarded)
    const float phi = -TWO_PI_OVER_256 * (float)(j - 31);
    float c0, s0, c1, s1, cs, ss;
    __sincosf(phi * (float)(2 * half), &s0, &c0);
    __sincosf(phi * (float)(2 * half + 1), &s1, &c1);
    __sincosf(phi * 4.0f, &ss, &cs);
    const float amp = 1.0f / 256.0f;
    c0 *= amp; s0 *= amp; c1 *= amp; s1 *= amp;

    const int colP = (part == 0) ? nh * 16 : 32 + nh * 16;
    const int colQ = (part == 0) ? 32 + nh * 16 : nh * 16;

    v8f accP = v8f_zero(), accQ = v8f_zero();
    for (int k0 = 0; k0 < 256; k0 += 4) {
      v2f aR, aI, bP, bQ;
      aR.x = c0; aR.y = c1;
      aI.x = s0; aI.y = s1;
      const float* r0 = &Xs[(k0 + 2 * half) * 64];
      bP.x = r0[colP + l]; bP.y = r0[64 + colP + l];
      bQ.x = r0[colQ + l]; bQ.y = r0[64 + colQ + l];
      accP = wmma_f32(aR, bP, accP);
      accQ = wmma_f32(aI, bQ, accQ);
      float tt;
      tt = c0 * cs - s0 * ss; s0 = c0 * ss + s0 * cs; c0 = tt;
      tt = c1 * cs - s1 * ss; s1 = c1 * ss + s1 * cs; c1 = tt;
    }
    float* plane = part ? xftI : xftR;
    const int kxcol = nh * 16 + l;
#pragma unroll
    for (int jj = 0; jj < 8; ++jj) {
      const int row = mtile * 16 + jj + 8 * half;
      const float val = part ? (accP[jj] + accQ[jj]) : (accP[jj] - accQ[jj]);
      if (row < 63) plane[((long)bi * 64 + row) * 32 + kxcol] = val;
    }
  }
}

// ---------------------------------------------------------------------------
// Kernel 3: per-mode complex channel mix: out[b,o] = sum_i x[b,i] * W[i,o].
// M=16(batch), K=64(Ci), N=64(Co). 4 real WMMA chains: RR-II, RI+IR.
// Weights assembled in LDS (col0 = [y0 | w00 | conj(flip(y0))], cols>0 = yposx).
// ---------------------------------------------------------------------------
__global__ __launch_bounds__(256) void k_modes(
    const float* __restrict__ xftR, const float* __restrict__ xftI,
    const float* __restrict__ y0r, const float* __restrict__ y0i,
    const float* __restrict__ ypr, const float* __restrict__ ypi,
    const float* __restrict__ w00, float* __restrict__ om) {
  __shared__ float Ar[16 * 64], Ai[16 * 64];
  __shared__ float Wr[64 * 64], Wi[64 * 64];
  const int tid = threadIdx.x;
  const int wave = tid >> 5, lane = tid & 31;
  const int half = lane >> 4, l = lane & 15;
  const int j = blockIdx.x >> 5;   // 0..62
  const int kx = blockIdx.x & 31;  // 0..31

  // async per-lane gather of the mode slice of x_ft (strided in i)
  for (int idx = tid; idx < 1024; idx += 256) {
    const int b = idx >> 6, i = idx & 63;
    const long off = ((long)(b * 64 + i) * 64 + j) * 32 + kx;
    async_ld_b32(lds_off(&Ar[idx]), xftR + off);
    async_ld_b32(lds_off(&Ai[idx]), xftI + off);
  }
  for (int idx = tid; idx < 4096; idx += 256) {
    const int i = idx >> 6, o = idx & 63;
    float wr, wi;
    if (kx > 0) {
      const long off = ((long)(i * 64 + o) * 63 + j) * 31 + (kx - 1);
      wr = ypr[off]; wi = ypi[off];
    } else if (j < 31) {
      const long off = (long)(i * 64 + o) * 31 + j;
      wr = y0r[off]; wi = y0i[off];
    } else if (j == 31) {
      wr = w00[i * 64 + o]; wi = 0.0f;
    } else {  // conj(flip(y0)): flipped index = 62 - j
      const long off = (long)(i * 64 + o) * 31 + (62 - j);
      wr = y0r[off]; wi = -y0i[off];
    }
    Wr[idx] = wr; Wi[idx] = wi;
  }
  wait_async0();
  __syncthreads();

  const int part = wave >> 2;   // 0 -> real output, 1 -> imag output
  const int ntile = wave & 3;   // Co tile
  v8f acc1 = v8f_zero(), acc2 = v8f_zero();
  for (int k0 = 0; k0 < 64; k0 += 4) {
    const int ka = k0 + 2 * half;
    v2f ar, ai, b1, b2;
    ar.x = Ar[l * 64 + ka]; ar.y = Ar[l * 64 + ka + 1];
    ai.x = Ai[l * 64 + ka]; ai.y = Ai[l * 64 + ka + 1];
    const int r0 = ka * 64 + ntile * 16 + l;
    if (part == 0) {
      b1.x = Wr[r0]; b1.y = Wr[r0 + 64];
      b2.x = Wi[r0]; b2.y = Wi[r0 + 64];
    } else {
      b1.x = Wi[r0]; b1.y = Wi[r0 + 64];
      b2.x = Wr[r0]; b2.y = Wr[r0 + 64];
    }
    acc1 = wmma_f32(ar, b1, acc1);  // part0: Ar*Wr   part1: Ar*Wi
    acc2 = wmma_f32(ai, b2, acc2);  // part0: Ai*Wi   part1: Ai*Wr
  }
  const int o = ntile * 16 + l;
  const int col = (part ? 32 : 0) + kx;
#pragma unroll
  for (int jj = 0; jj < 8; ++jj) {
    const int b = jj + 8 * half;
    const float val = part ? (acc1[jj] + acc2[jj]) : (acc1[jj] - acc2[jj]);
    om[(((long)(b * 64 + o)) * 64 + j) * 64 + col] = val;
  }
}

// ---------------------------------------------------------------------------
// Kernel 4: inverse column transform: 63 modes -> 256 spatial rows (complex).
// Per (b,o): G(256 x 32 cplx) = E(256 x 63 cplx twiddle) x M(63 x 32 cplx).
// Inverse ortho scale 1/256 folded into E. Padding row j=63 zeroed in LDS.
// ---------------------------------------------------------------------------
__global__ __launch_bounds__(256) void k_icol(const float* __restrict__ om,
                                              float* __restrict__ Gcol) {
  __shared__ __align__(16) float Bs[64 * 64];
  const int tid = threadIdx.x;
  const int wave = tid >> 5, lane = tid & 31;
  const int half = lane >> 4, l = lane & 15;
  const int bo = blockIdx.x;  // b*64 + o

  const float* src = om + (long)bo * 64 * 64;
  for (int idx = tid * 4; idx < 63 * 64; idx += 1024)
    async_ld_b128(lds_off(&Bs[idx]), src + idx);
  if (tid < 16) {  // zero the j==63 padding row directly in LDS
    float4 z; z.x = 0.f; z.y = 0.f; z.z = 0.f; z.w = 0.f;
    *(float4*)&Bs[63 * 64 + tid * 4] = z;
  }
  wait_async0();
  __syncthreads();

  float* dst = Gcol + (long)bo * 256 * 64;
  for (int mt = wave; mt < 16; mt += 8) {
    const int u = mt * 16 + l;
    const float phi = TWO_PI_OVER_256 * (float)u;  // +i direction (inverse)
    float c0, s0, c1, s1, cs, ss;
    __sincosf(phi * (float)(2 * half - 31), &s0, &c0);
    __sincosf(phi * (float)(2 * half - 30), &s1, &c1);
    __sincosf(phi * 4.0f, &ss, &cs);
    const float amp = 1.0f / 256.0f;
    c0 *= amp; s0 *= amp; c1 *= amp; s1 *= amp;

    v8f P[4], Q[4];
#pragma unroll
    for (int nn = 0; nn < 4; ++nn) { P[nn] = v8f_zero(); Q[nn] = v8f_zero(); }

    for (int k0 = 0; k0 < 64; k0 += 4) {
      v2f aR, aI;
      aR.x = c0; aR.y = c1;
      aI.x = s0; aI.y = s1;
      if (k0 == 60 && half == 1) { aR.y = 0.0f; aI.y = 0.0f; }  // j==63 pad
      const float* r0 = &Bs[(k0 + 2 * half) * 64 + l];
#pragma unroll
      for (int nn = 0; nn < 4; ++nn) {
        v2f b; b.x = r0[nn * 16]; b.y = r0[64 + nn * 16];
        P[nn] = wmma_f32(aR, b, P[nn]);  // Er * [Mr|Mi]
        Q[nn] = wmma_f32(aI, b, Q[nn]);  // Ei * [Mr|Mi]
      }
      float tt;
      tt = c0 * cs - s0 * ss; s0 = c0 * ss + s0 * cs; c0 = tt;
      tt = c1 * cs - s1 * ss; s1 = c1 * ss + s1 * cs; c1 = tt;
    }
#pragma unroll
    for (int nn = 0; nn < 2; ++nn) {
#pragma unroll
      for (int jj = 0; jj < 8; ++jj) {
        const int row = mt * 16 + jj + 8 * half;
        dst[(long)row * 64 + nn * 16 + l]      = P[nn][jj] - Q[nn + 2][jj];
        dst[(long)row * 64 + 32 + nn * 16 + l] = P[nn + 2][jj] + Q[nn][jj];
      }
    }
  }
}

// ---------------------------------------------------------------------------
// Kernel 5: inverse row transform to real output (Hermitian fold):
// y[u,v] = sum_kx c(kx) * (gr*cos(2pi kx v/256) - gi*sin(2pi kx v/256)),
// c(0)=1, c(kx>0)=2. M=64 rows per block, K=64 ([gr|gi]), N=256.
// ---------------------------------------------------------------------------
__global__ __launch_bounds__(256) void k_irow(const float* __restrict__ Gcol,
                                              float* __restrict__ y) {
  __shared__ __align__(16) float As[64 * 68];  // padded stride 68
  const int tid = threadIdx.x;
  const int wave = tid >> 5, lane = tid & 31;
  const int half = lane >> 4, l = lane & 15;
  const int bo = blockIdx.x >> 2;
  const int grp = blockIdx.x & 3;  // 64-row group of u

  const float* src = Gcol + ((long)bo * 256 + grp * 64) * 64;
  for (int idx = tid * 4; idx < 64 * 64; idx += 1024) {
    const int r = idx >> 6, c = idx & 63;
    async_ld_b128(lds_off(&As[r * 68 + c]), src + idx);
  }
  wait_async0();
  __syncthreads();

  const int mt = wave & 3;   // u tile within group
  const int nh = wave >> 2;  // which 128-wide half of v

  float ic0[8], is0[8], ic1[8], is1[8], stc[8], sts[8];
  float zc0[8], zs0[8], zc1[8], zs1[8];
#pragma unroll
  for (int nn = 0; nn < 8; ++nn) {
    const int v = nh * 128 + nn * 16 + l;
    const float phi = TWO_PI_OVER_256 * (float)v;
    __sincosf(phi * (float)(2 * half), &is0[nn], &ic0[nn]);
    __sincosf(phi * (float)(2 * half + 1), &is1[nn], &ic1[nn]);
    __sincosf(phi * 4.0f, &sts[nn], &stc[nn]);
    zc0[nn] = ic0[nn]; zs0[nn] = is0[nn];
    zc1[nn] = ic1[nn]; zs1[nn] = is1[nn];
  }

  v8f acc[8];
#pragma unroll
  for (int nn = 0; nn < 8; ++nn) acc[nn] = v8f_zero();

  const float* arow = &As[(mt * 16 + l) * 68];
  for (int ks = 0; ks < 16; ++ks) {
    const int k0 = ks * 4;
    const bool sinPart = (ks >= 8);
    if (ks == 8) {  // kx restarts for the gi half: reset rotations
#pragma unroll
      for (int nn = 0; nn < 8; ++nn) {
        zc0[nn] = ic0[nn]; zs0[nn] = is0[nn];
        zc1[nn] = ic1[nn]; zs1[nn] = is1[nn];
      }
    }
    v2f a;
    a.x = arow[k0 + 2 * half];
    a.y = arow[k0 + 2 * half + 1];
    const int kxa = (k0 & 31) + 2 * half;
    const float f0 = (kxa == 0) ? 1.0f : 2.0f;
#pragma unroll
    for (int nn = 0; nn < 8; ++nn) {
      v2f b;
      if (sinPart) { b.x = -f0 * zs0[nn]; b.y = -2.0f * zs1[nn]; }
      else         { b.x =  f0 * zc0[nn]; b.y =  2.0f * zc1[nn]; }
      acc[nn] = wmma_f32(a, b, acc[nn]);
      float tt;
      tt = zc0[nn] * stc[nn] - zs0[nn] * sts[nn];
      zs0[nn] = zc0[nn] * sts[nn] + zs0[nn] * stc[nn];
      zc0[nn] = tt;
      tt = zc1[nn] * stc[nn] - zs1[nn] * sts[nn];
      zs1[nn] = zc1[nn] * sts[nn] + zs1[nn] * stc[nn];
      zc1[nn] = tt;
    }
  }

  float* dst = y + ((long)bo * 256 + grp * 64 + mt * 16 + 8 * half) * 256 +
               nh * 128 + l;
#pragma unroll
  for (int nn = 0; nn < 8; ++nn) {
#pragma unroll
    for (int jj = 0; jj < 8; ++jj)
      dst[(long)jj * 256 + nn * 16] = acc[nn][jj];
  }
}

// ---------------------------------------------------------------------------
extern "C" void kernel_launch(void* const* d_in, const int* in_sizes, int n_in,
                              void* d_out, int out_size, void* d_ws,
                              size_t ws_size, hipStream_t stream) {
  (void)in_sizes; (void)n_in; (void)out_size; (void)ws_size;
  const float* x   = (const float*)d_in[0];
  const float* y0r = (const float*)d_in[1];
  const float* y0i = (const float*)d_in[2];
  const float* ypr = (const float*)d_in[3];
  const float* ypi = (const float*)d_in[4];
  const float* w00 = (const float*)d_in[5];
  float* out = (float*)d_out;

  // Workspace partition (floats): total 41,943,040 f = 160 MiB
  float* Xrow = (float*)d_ws;                    // [262144][64]  (R|I)
  float* xftR = Xrow + (size_t)262144 * 64;      // [1024][64][32]
  float* xftI = xftR + (size_t)1024 * 64 * 32;   // [1024][64][32]
  float* om   = xftI + (size_t)1024 * 64 * 32;   // [1024][64][64] (R|I)
  float* Gcol = om + (size_t)1024 * 64 * 64;     // [1024][256][64] (R|I)

  k_row_dft<<<8192, 256, 0, stream>>>(x, Xrow);
  k_col_dft<<<1024, 256, 0, stream>>>(Xrow, xftR, xftI);
  k_modes  <<<2016, 256, 0, stream>>>(xftR, xftI, y0r, y0i, ypr, ypi, w00, om);
  k_icol   <<<1024, 256, 0, stream>>>(om, Gcol);
  k_irow   <<<4096, 256, 0, stream>>>(Gcol, out);
}